// ModulatedDeformConvWithOff_11982958756209
// MI455X (gfx1250) — compile-verified
//
#include <hip/hip_runtime.h>
#include <math.h>

typedef __attribute__((ext_vector_type(2))) float v2f;
typedef __attribute__((ext_vector_type(8))) float v8f;

#define C_IN   256
#define H_IN   64
#define W_IN   64
#define C_OUT  256
#define KKTAPS 9
#define TILE_P 16   // output pixels per block (one row segment)
#define CB     64   // reduction channels per chunk
#define NCH    (KKTAPS * (C_IN / CB))   // 36 chunks

// ---------------------------------------------------------------------------
// One-shot weight transpose: w_dcn (Cout, C, 3, 3) -> interleaved
//   wT[((kk*(C/4) + c/4) * Cout + o) * 4 + (c%4)]
// so a WMMA B-fragment (K=4 consecutive c, 16 consecutive o) is a single
// coalesced global_load_b64 per lane (16B lane stride).
// ---------------------------------------------------------------------------
__global__ __launch_bounds__(256)
void wdcn_transpose_kernel(const float* __restrict__ w, float* __restrict__ wt)
{
    int i = blockIdx.x * 256 + threadIdx.x;     // 9*64*256*4 = 589824 exact
    int cl = i & 3;
    int o  = (i >> 2) & 255;
    int cg = (i >> 10) & 63;
    int kk = i >> 16;
    int c  = cg * 4 + cl;
    wt[i] = w[(size_t)o * (C_IN * KKTAPS) + c * KKTAPS + kk];
}

// ---------------------------------------------------------------------------
// Fused DCNv2 forward.
//  Phase A: stage 3x18 x-patch in LDS, 27-ch offset conv (VALU), sigmoid,
//           precompute per-(tap,pixel) clamped corner offsets + blend weights.
//  Phase B: software-pipelined: gathers for chunk i+1 issue before the 32
//           f32 WMMAs of chunk i (double-buffered LDS v-tile, 1 barrier/chunk).
// ---------------------------------------------------------------------------
template <bool USE_WT>
__global__ __launch_bounds__(256)
void dcnv2_fused_kernel(const float* __restrict__ x,
                        const float* __restrict__ w_off,
                        const float* __restrict__ b_off,
                        const float* __restrict__ w_dcn,
                        const float* __restrict__ wT,
                        const float* __restrict__ b_dcn,
                        float* __restrict__ out)
{
    // union region: phase A x-patch (55296B) / phase B double v-buffer (8192B)
    __shared__ __align__(16) float s_big[C_IN * 54];
    __shared__ float s_om[27 * TILE_P];                //  1728 B
    __shared__ int   s_poff[KKTAPS * TILE_P * 4];      //  2304 B
    __shared__ float s_pw[KKTAPS * TILE_P * 4];        //  2304 B

    const int t    = threadIdx.x;
    const int wv   = t >> 5;
    const int lane = t & 31;

    const int bi  = blockIdx.x;
    const int b   = bi >> 8;
    const int rem = bi & 255;
    const int h   = rem >> 2;           // Ho = 64
    const int w0  = (rem & 3) << 4;     // 4 tiles of 16 across Wo = 64

    const float* xb = x + (size_t)b * C_IN * H_IN * W_IN;

    // ------------- Phase A: stage x patch (rows h-1..h+1, cols w0-1..w0+16)
    for (int i = t; i < C_IN * 54; i += 256) {
        int c   = i / 54;
        int r   = (i % 54) / 18;
        int col = i % 18;
        int gy = h - 1 + r;
        int gx = w0 - 1 + col;
        float v = 0.0f;
        if (gy >= 0 && gy < H_IN && gx >= 0 && gx < W_IN)
            v = xb[(size_t)c * (H_IN * W_IN) + gy * W_IN + gx];
        s_big[i] = v;
    }
    __syncthreads();

    // ------------- offset conv: 27 channels x 16 pixels, from LDS patch
    for (int idx = t; idx < 27 * TILE_P; idx += 256) {
        int j = idx >> 4;
        int p = idx & 15;
        float acc = b_off[j];
        const float* wj = w_off + (size_t)j * (C_IN * 9);
        for (int c = 0; c < C_IN; ++c) {
            const float* xs = &s_big[c * 54];
            const float* wc = &wj[c * 9];
            #pragma unroll
            for (int ky = 0; ky < 3; ++ky)
                #pragma unroll
                for (int kx = 0; kx < 3; ++kx)
                    acc = fmaf(xs[ky * 18 + p + kx], wc[ky * 3 + kx], acc);
        }
        s_om[j * TILE_P + p] = acc;
    }
    __syncthreads();

    // ------------- bilinear params per (tap kk, pixel p):
    //   4 clamped plane offsets + 4 blend weights (validity & mask folded)
    if (t < KKTAPS * TILE_P) {
        int kk = t >> 4;
        int p  = t & 15;
        float offy = s_om[(2 * kk) * TILE_P + p];
        float offx = s_om[(2 * kk + 1) * TILE_P + p];
        float mz   = s_om[(18 + kk) * TILE_P + p];
        float m    = 1.0f / (1.0f + __expf(-mz));           // sigmoid
        float sy = (float)(h - 1 + kk / 3) + offy;          // stride1, pad1, dil1
        float sx = (float)(w0 + p - 1 + kk % 3) + offx;
        float y0f = floorf(sy);
        float x0f = floorf(sx);
        float wy = sy - y0f, wx = sx - x0f;
        int iy0 = (int)y0f, ix0 = (int)x0f;
        int iy1 = iy0 + 1,  ix1 = ix0 + 1;
        int cy0 = min(max(iy0, 0), H_IN - 1);
        int cy1 = min(max(iy1, 0), H_IN - 1);
        int cx0 = min(max(ix0, 0), W_IN - 1);
        int cx1 = min(max(ix1, 0), W_IN - 1);
        float vy0 = (iy0 >= 0 && iy0 < H_IN) ? m : 0.0f;
        float vy1 = (iy1 >= 0 && iy1 < H_IN) ? m : 0.0f;
        float vx0 = (ix0 >= 0 && ix0 < W_IN) ? 1.0f : 0.0f;
        float vx1 = (ix1 >= 0 && ix1 < W_IN) ? 1.0f : 0.0f;
        int*   po = &s_poff[t * 4];
        float* pw = &s_pw[t * 4];
        po[0] = cy0 * W_IN + cx0;  pw[0] = (1.0f - wy) * (1.0f - wx) * vy0 * vx0;
        po[1] = cy0 * W_IN + cx1;  pw[1] = (1.0f - wy) * wx          * vy0 * vx1;
        po[2] = cy1 * W_IN + cx0;  pw[2] = wy          * (1.0f - wx) * vy1 * vx0;
        po[3] = cy1 * W_IN + cx1;  pw[3] = wy          * wx          * vy1 * vx1;
    }
    __syncthreads();

    // ------------- Phase B: pipelined deformable GEMM via f32 WMMA 16x16x4
    v8f acc0 = {};
    v8f acc1 = {};
    const int obase = wv * 32;                 // 2 N-tiles per wave
    const int koff  = (lane < 16) ? 0 : 2;     // f32 fragment K split
    const int nlane = lane & 15;
    const int prow  = lane & 15;               // A-matrix M = lane%16
    const int myp   = t & 15;                  // im2col pixel (fixed per thread)
    const int mycc0 = t >> 4;                  // im2col channel base (0..15)

    // prologue: im2col chunk 0 -> buffer 0
    {
        const int pp = (0 * TILE_P + myp) * 4;
        const int   o0 = s_poff[pp + 0], o1 = s_poff[pp + 1],
                    o2 = s_poff[pp + 2], o3 = s_poff[pp + 3];
        const float g0 = s_pw[pp + 0],  g1 = s_pw[pp + 1],
                    g2 = s_pw[pp + 2],  g3 = s_pw[pp + 3];
        float fA[4], fB[4], fC[4], fD[4];
        #pragma unroll
        for (int e = 0; e < 4; ++e) {
            const float* base = xb + (size_t)(mycc0 + e * 16) * (H_IN * W_IN);
            fA[e] = base[o0]; fB[e] = base[o1]; fC[e] = base[o2]; fD[e] = base[o3];
        }
        #pragma unroll
        for (int e = 0; e < 4; ++e)
            s_big[myp * CB + mycc0 + e * 16] =
                fA[e] * g0 + fB[e] * g1 + fC[e] * g2 + fD[e] * g3;
    }
    __syncthreads();

    for (int i = 0; i < NCH; ++i) {
        const float* cur = s_big + (i & 1) * (TILE_P * CB);
        float*       nxt = s_big + ((i + 1) & 1) * (TILE_P * CB);
        const int kk = i >> 2;
        const int c0 = (i & 3) * CB;

        // --- issue gathers for chunk i+1 (overlap with WMMA below)
        float fA[4], fB[4], fC[4], fD[4];
        float g0 = 0, g1 = 0, g2 = 0, g3 = 0;
        const bool has_next = (i + 1) < NCH;
        if (has_next) {
            const int nkk = (i + 1) >> 2;
            const int nc0 = ((i + 1) & 3) * CB;
            const int pp = (nkk * TILE_P + myp) * 4;
            const int o0 = s_poff[pp + 0], o1 = s_poff[pp + 1],
                      o2 = s_poff[pp + 2], o3 = s_poff[pp + 3];
            g0 = s_pw[pp + 0]; g1 = s_pw[pp + 1];
            g2 = s_pw[pp + 2]; g3 = s_pw[pp + 3];
            #pragma unroll
            for (int e = 0; e < 4; ++e) {
                const float* base = xb + (size_t)(nc0 + mycc0 + e * 16) * (H_IN * W_IN);
                fA[e] = base[o0]; fB[e] = base[o1];
                fC[e] = base[o2]; fD[e] = base[o3];
            }
        }

        // --- 16 K-steps x 2 N-tiles of f32 WMMA on chunk i
        #pragma unroll
        for (int ks = 0; ks < CB / 4; ++ks) {
            int kbase = ks * 4 + koff;
            v2f a;
            a.x = cur[prow * CB + kbase];
            a.y = cur[prow * CB + kbase + 1];
            v2f bf0, bf1;
            if (USE_WT) {
                int cg = (c0 >> 2) + ks;
                const float* wbase = wT +
                    ((size_t)(kk * (C_IN / 4) + cg) * C_OUT) * 4 + koff;
                bf0 = *(const v2f*)(wbase + (size_t)(obase + nlane) * 4);
                bf1 = *(const v2f*)(wbase + (size_t)(obase + 16 + nlane) * 4);
            } else {
                int c = c0 + kbase;
                const float* wp0 = w_dcn + (size_t)(obase + nlane) * (C_IN * KKTAPS)
                                         + (size_t)c * KKTAPS + kk;
                bf0.x = wp0[0]; bf0.y = wp0[KKTAPS];
                const float* wp1 = w_dcn + (size_t)(obase + 16 + nlane) * (C_IN * KKTAPS)
                                         + (size_t)c * KKTAPS + kk;
                bf1.x = wp1[0]; bf1.y = wp1[KKTAPS];
            }
            acc0 = __builtin_amdgcn_wmma_f32_16x16x4_f32(
                    false, a, false, bf0, (short)0, acc0, false, false);
            acc1 = __builtin_amdgcn_wmma_f32_16x16x4_f32(
                    false, a, false, bf1, (short)0, acc1, false, false);
        }

        // --- blend + store chunk i+1 tile
        if (has_next) {
            #pragma unroll
            for (int e = 0; e < 4; ++e)
                nxt[myp * CB + mycc0 + e * 16] =
                    fA[e] * g0 + fB[e] * g1 + fC[e] * g2 + fD[e] * g3;
        }
        __syncthreads();
    }

    // ------------- epilogue: scatter D tiles + bias
    float* outb = out + (size_t)b * C_OUT * (H_IN * W_IN) + h * W_IN + w0;
    const int mhi = (lane < 16) ? 0 : 8;
    {
        int o = obase + nlane;
        float bias = b_dcn[o];
        #pragma unroll
        for (int r = 0; r < 8; ++r)
            outb[(size_t)o * (H_IN * W_IN) + (r + mhi)] = acc0[r] + bias;
        o = obase + 16 + nlane;
        bias = b_dcn[o];
        #pragma unroll
        for (int r = 0; r < 8; ++r)
            outb[(size_t)o * (H_IN * W_IN) + (r + mhi)] = acc1[r] + bias;
    }
}

extern "C" void kernel_launch(void* const* d_in, const int* in_sizes, int n_in,
                              void* d_out, int out_size, void* d_ws, size_t ws_size,
                              hipStream_t stream) {
    (void)in_sizes; (void)n_in; (void)out_size;
    const float* x     = (const float*)d_in[0];
    const float* w_off = (const float*)d_in[1];
    const float* b_off = (const float*)d_in[2];
    const float* w_dcn = (const float*)d_in[3];
    const float* b_dcn = (const float*)d_in[4];
    float* out = (float*)d_out;

    const size_t wt_bytes = (size_t)C_OUT * C_IN * KKTAPS * sizeof(float);
    dim3 grid(8 * 64 * 4);   // B=8 images, 64 rows, 4 tiles of 16 pixels
    dim3 block(256);

    if (ws_size >= wt_bytes) {
        float* wT = (float*)d_ws;
        wdcn_transpose_kernel<<<dim3((C_OUT * C_IN * KKTAPS) / 256), block, 0, stream>>>(
            w_dcn, wT);
        dcnv2_fused_kernel<true><<<grid, block, 0, stream>>>(
            x, w_off, b_off, w_dcn, wT, b_dcn, out);
    } else {
        dcnv2_fused_kernel<false><<<grid, block, 0, stream>>>(
            x, w_off, b_off, w_dcn, w_dcn, b_dcn, out);
    }
}